// DynFilter_8942121910817
// MI455X (gfx1250) — compile-verified
//
#include <hip/hip_runtime.h>

// Problem sizes (fixed by the reference).
#define BB 8
#define CC 3
#define HH 544
#define WW 960
#define KK 27            // C*KH*KW
#define NTHREADS 256
#define PXT 4            // pixels per thread
#define ACTIVE (WW / PXT)   // 240 compute threads

typedef __attribute__((ext_vector_type(4))) float f4;
typedef __attribute__((ext_vector_type(4))) int   v4i;

// LDS tile: 9 rows (c*3+kh), each holds x[hy][w=-1..960] with w=0 at float
// index 4 so interior B128 async writes are 16B aligned. Stride 968 floats
// keeps every row base 16B aligned (968*4 = 3872 = 242*16).
#define ROWS 9
#define RSTRIDE 968

#if __has_builtin(__builtin_amdgcn_global_load_async_to_lds_b128)
#define ASYNC_STAGE 1
#else
#define ASYNC_STAGE 0
#endif

__device__ __forceinline__ void stage_b128(const float* g, float* l) {
#if ASYNC_STAGE
  // CDNA5 async DMA: 16B per lane, global -> LDS, tracked by ASYNCcnt.
  // clang-22 signature (from round-1 diagnostic): param0 is
  // 'v4i32 __device__ *' (= addrspace(1) pointer to 4xi32), so cast
  // through generic first, then addrspace-cast.
  __builtin_amdgcn_global_load_async_to_lds_b128(
      (__attribute__((address_space(1))) v4i*)(v4i*)(void*)g,
      (__attribute__((address_space(3))) v4i*)(v4i*)(void*)l,
      /*offset=*/0, /*cpol=*/0);
#else
  *(f4*)l = *(const f4*)g;   // synchronous fallback
#endif
}

__device__ __forceinline__ void wait_async0() {
#if ASYNC_STAGE
#if __has_builtin(__builtin_amdgcn_s_wait_asynccnt)
  __builtin_amdgcn_s_wait_asynccnt(0);
#else
  asm volatile("s_wait_asynccnt 0" ::: "memory");
#endif
#endif
}

__global__ __launch_bounds__(NTHREADS)
void dynfilter_kernel(const float* __restrict__ x,
                      const float* __restrict__ filt,
                      float* __restrict__ out) {
  __shared__ alignas(16) float xs[ROWS * RSTRIDE];

  const int tid = threadIdx.x;
  const int bid = blockIdx.x;          // one block per (b, h) output row
  const int b   = bid / HH;
  const int h   = bid % HH;

  // ---- 1) Zero only the pad cells / out-of-bounds rows (ds stores).
  //      Addresses are disjoint from the async-written interior, so no
  //      DScnt-vs-ASYNCcnt ordering hazard exists.
  for (int r = 0; r < ROWS; ++r) {
    const int hy = h + (r % 3) - 1;    // uniform per block
    if (hy < 0 || hy >= HH) {
      for (int i = tid; i < 962; i += NTHREADS) xs[r * RSTRIDE + 3 + i] = 0.0f;
    } else if (tid == 0) {
      xs[r * RSTRIDE + 3]   = 0.0f;    // w = -1
      xs[r * RSTRIDE + 964] = 0.0f;    // w = 960
    }
  }

  // ---- 2) Async-stage the 9 x-rows (regular temporal: x is L2-resident
  //      and reused 9x; the NT filter stream below won't evict it).
  if (tid < ACTIVE) {
    for (int r = 0; r < ROWS; ++r) {
      const int c  = r / 3;
      const int hy = h + (r % 3) - 1;
      if (hy < 0 || hy >= HH) continue;
      const float* src = x + (((size_t)(b * CC + c) * HH + hy) * WW) + 4 * tid;
      float*       dst = &xs[r * RSTRIDE + 4 + 4 * tid];
      stage_b128(src, dst);
    }
  }
  wait_async0();        // own wave's ASYNCcnt -> 0
  __syncthreads();      // dscnt drain + barrier: whole tile visible

  // ---- 3) Compute: 27 NT b128 filter loads + 27 b128 LDS loads + 108 FMAs
  //      per thread for a float4 of outputs.
  if (tid < ACTIVE) {
    const int w0 = tid * PXT;
    const size_t fplane = (size_t)HH * WW;
    const float* fbase = filt + (size_t)b * KK * fplane + (size_t)h * WW + w0;

    f4 acc = (f4){0.0f, 0.0f, 0.0f, 0.0f};

#pragma unroll
    for (int r = 0; r < ROWS; ++r) {
      const float* row = &xs[r * RSTRIDE + w0];     // 16B aligned
      const f4 a  = *(const f4*)(row);              // idx w0-4 .. w0-1
      const f4 m  = *(const f4*)(row + 4);          // idx w0   .. w0+3
      const f4 cq = *(const f4*)(row + 8);          // idx w0+4 .. w0+7

      const f4 t0 = (f4){a.w, m.x, m.y, m.z};       // taps, kw = 0 (w-1..w+2)
      const f4 t1 = m;                              // taps, kw = 1 (w  ..w+3)
      const f4 t2 = (f4){m.y, m.z, m.w, cq.x};      // taps, kw = 2 (w+1..w+4)

      const int k = r * 3;                          // c*9 + kh*3 (+ kw)
      const f4 f0 = __builtin_nontemporal_load((const f4*)(fbase + (size_t)(k + 0) * fplane));
      const f4 f1 = __builtin_nontemporal_load((const f4*)(fbase + (size_t)(k + 1) * fplane));
      const f4 f2 = __builtin_nontemporal_load((const f4*)(fbase + (size_t)(k + 2) * fplane));

      acc += t0 * f0;
      acc += t1 * f1;
      acc += t2 * f2;
    }

    float* optr = out + ((size_t)b * HH + h) * WW + w0;
    __builtin_nontemporal_store(acc, (f4*)optr);    // touch-once output
  }
}

extern "C" void kernel_launch(void* const* d_in, const int* in_sizes, int n_in,
                              void* d_out, int out_size, void* d_ws, size_t ws_size,
                              hipStream_t stream) {
  const float* x    = (const float*)d_in[0];   // [8, 3, 544, 960] fp32
  const float* filt = (const float*)d_in[1];   // [8, 27, 544, 960] fp32
  float*       out  = (float*)d_out;           // [8, 1, 544, 960] fp32

  dim3 grid(BB * HH);                          // 4352 blocks, one per (b,h)
  dim3 block(NTHREADS);                        // 8 wave32 per block
  dynfilter_kernel<<<grid, block, 0, stream>>>(x, filt, out);
}